// Model_72499047956494
// MI455X (gfx1250) — compile-verified
//
#include <hip/hip_runtime.h>

typedef __attribute__((ext_vector_type(2))) float f32x2;
typedef __attribute__((ext_vector_type(8))) float f32x8;

#define HP 100000
#define HT 400000
#define HH 128
#define HE 2000000
#define HEL 500000
#define LDS_PITCH 140   // 560B row stride: 16B aligned (async b128 to LDS), bank step 12 -> conflict-free

// ---------------- zero workspace ----------------
__global__ void k_zero(float4* __restrict__ p, long n4) {
    long i = (long)blockIdx.x * blockDim.x + threadIdx.x;
    if (i < n4) p[i] = make_float4(0.f, 0.f, 0.f, 0.f);
}

// ---------------- degree count ----------------
__global__ void k_degree(const int* __restrict__ src, const int* __restrict__ dst,
                         float* __restrict__ degP, float* __restrict__ degT, int n) {
    int i = blockIdx.x * blockDim.x + threadIdx.x;
    if (i < n) {
        atomicAdd(&degT[dst[i]], 1.0f);
        atomicAdd(&degP[src[i]], 1.0f);
    }
}

__global__ void k_invdeg(float* __restrict__ d, int n) {
    int i = blockIdx.x * blockDim.x + threadIdx.x;
    if (i < n) d[i] = 1.0f / fmaxf(d[i], 1.0f);
}

// ---------------- edge scatter (both directions in one pass) ----------------
// one wave (32 lanes) per edge; each lane moves a float4 (H = 128 = 32*4)
__global__ void k_scatter(const float* __restrict__ xP,   // [P,H] features on src side
                          const float* __restrict__ xT,   // [T,H] features on dst side
                          const int* __restrict__ src, const int* __restrict__ dst,
                          float* __restrict__ aggT, float* __restrict__ aggP, int n) {
    long idx = (long)blockIdx.x * blockDim.x + threadIdx.x;
    int e = (int)(idx >> 5);
    int lane = (int)(idx & 31);
    if (e >= n) return;
    int s = src[e];
    int d = dst[e];
    const float4 fs = *(const float4*)(xP + (long)s * HH + lane * 4);
    float* at = aggT + (long)d * HH + lane * 4;
    atomicAdd(at + 0, fs.x); atomicAdd(at + 1, fs.y);
    atomicAdd(at + 2, fs.z); atomicAdd(at + 3, fs.w);
    const float4 fd = *(const float4*)(xT + (long)d * HH + lane * 4);
    float* ap = aggP + (long)s * HH + lane * 4;
    atomicAdd(ap + 0, fd.x); atomicAdd(ap + 1, fd.y);
    atomicAdd(ap + 2, fd.z); atomicAdd(ap + 3, fd.w);
}

// ---------------- SAGE layer: out = (agg/deg) @ Wl^T + bl + xdst @ Wr^T  (opt. relu) ----------------
// 256 threads = 8 waves; workgroup owns 16 rows x 128 cols; wave w owns cols [16w,16w+16).
// Tiles are staged RAW into LDS with GLOBAL_LOAD_ASYNC_TO_LDS_B128 (ASYNCcnt), then each
// lane folds 1/deg into its A rows at WMMA-consumption time (lane l only holds A-row l&15).
// Two independent V_WMMA_F32_16X16X4_F32 accumulator chains (Wl-GEMM, Wr-GEMM) for ILP.
// Safe with out == aggA (in-place): reads fully staged to LDS before the barrier.
__global__ __launch_bounds__(256) void k_sage_gemm(
    const float* __restrict__ aggA,   // [rows,H] neighbor sums
    const float* __restrict__ invdeg, // [rows]
    const float* __restrict__ xdst,   // [rows,H] self features
    const float* __restrict__ Wl, const float* __restrict__ bl,
    const float* __restrict__ Wr,
    float* __restrict__ out, int doRelu)
{
    __shared__ float ldsMean[16 * LDS_PITCH];
    __shared__ float ldsDst [16 * LDS_PITCH];

    const int tid = threadIdx.x;
    const long ibase = (long)blockIdx.x * 16;

    // async-stage 16x128 agg rows and 16x128 self rows directly into LDS
    #pragma unroll
    for (int i = tid; i < 512; i += 256) {
        int r = i >> 5;
        int c = (i & 31) << 2;
        long row = ibase + r;
        const float* gM = aggA + row * HH + c;
        const float* gD = xdst + row * HH + c;
        unsigned lM = (unsigned)(size_t)&ldsMean[r * LDS_PITCH + c];
        unsigned lD = (unsigned)(size_t)&ldsDst [r * LDS_PITCH + c];
        asm volatile("global_load_async_to_lds_b128 %0, %1, off"
                     :: "v"(lM), "v"(gM) : "memory");
        asm volatile("global_load_async_to_lds_b128 %0, %1, off"
                     :: "v"(lD), "v"(gD) : "memory");
    }
    asm volatile("s_wait_asynccnt 0x0" ::: "memory");
    __syncthreads();

    const int wave = tid >> 5;          // 0..7 -> column tile
    const int lane = tid & 31;
    const int m    = lane & 15;         // A row / B,C column within tile
    const int kh   = (lane >> 4) << 1;  // K sub-offset 0 or 2
    const int jbase = wave << 4;

    const float* pM  = &ldsMean[m * LDS_PITCH + kh];
    const float* pD  = &ldsDst [m * LDS_PITCH + kh];
    const float* pWl = Wl + (long)(jbase + m) * HH + kh;  // B = W^T: contiguous along K
    const float* pWr = Wr + (long)(jbase + m) * HH + kh;

    const float idg = invdeg[ibase + m];  // lane's A row is fixed -> single scalar

    f32x8 accL = {0.f, 0.f, 0.f, 0.f, 0.f, 0.f, 0.f, 0.f};
    f32x8 accR = {0.f, 0.f, 0.f, 0.f, 0.f, 0.f, 0.f, 0.f};
    #pragma unroll
    for (int k = 0; k < HH; k += 4) {
        f32x2 a1 = *(const f32x2*)(pM + k);
        a1 = a1 * idg;                                    // mean = agg/deg, folded here
        f32x2 b1 = *(const f32x2*)(pWl + k);
        accL = __builtin_amdgcn_wmma_f32_16x16x4_f32(false, a1, false, b1,
                                                     (short)0, accL, false, false);
        f32x2 a2 = *(const f32x2*)(pD + k);
        f32x2 b2 = *(const f32x2*)(pWr + k);
        accR = __builtin_amdgcn_wmma_f32_16x16x4_f32(false, a2, false, b2,
                                                     (short)0, accR, false, false);
    }

    const float bv = bl[jbase + m];
    const int rowAdd = (lane >> 4) << 3;  // upper lane half holds rows +8
    const long col = jbase + m;
    #pragma unroll
    for (int r = 0; r < 8; ++r) {
        float v = accL[r] + accR[r] + bv;
        if (doRelu) v = fmaxf(v, 0.f);
        out[(ibase + r + rowAdd) * HH + col] = v;
    }
}

// ---------------- dot-product decoder on label edges ----------------
__global__ void k_dot(const float* __restrict__ p2, const float* __restrict__ t2,
                      const int* __restrict__ lsrc, const int* __restrict__ ldst,
                      float* __restrict__ out, int n) {
    long idx = (long)blockIdx.x * blockDim.x + threadIdx.x;
    int e = (int)(idx >> 5);
    int lane = (int)(idx & 31);
    if (e >= n) return;
    int a = lsrc[e];
    int b = ldst[e];
    float4 fa = *(const float4*)(p2 + (long)a * HH + lane * 4);
    float4 fb = *(const float4*)(t2 + (long)b * HH + lane * 4);
    float s = fa.x * fb.x + fa.y * fb.y + fa.z * fb.z + fa.w * fb.w;
    #pragma unroll
    for (int off = 16; off > 0; off >>= 1) s += __shfl_xor(s, off, 32);
    if (lane == 0) out[e] = s;
}

extern "C" void kernel_launch(void* const* d_in, const int* in_sizes, int n_in,
                              void* d_out, int out_size, void* d_ws, size_t ws_size,
                              hipStream_t stream) {
    (void)in_sizes; (void)n_in; (void)out_size; (void)ws_size;
    // inputs (node_id_* are identity arange -> ignored)
    const float* xp   = (const float*)d_in[2];   // [P,H]
    const float* xt   = (const float*)d_in[3];   // [T,H]
    const int*  esrc  = (const int*)d_in[4];
    const int*  edst  = (const int*)d_in[5];
    const int*  lsrc  = (const int*)d_in[6];
    const int*  ldst  = (const int*)d_in[7];
    const float* c1h_Wl = (const float*)d_in[8],  *c1h_bl = (const float*)d_in[9],  *c1h_Wr = (const float*)d_in[10];
    const float* c1r_Wl = (const float*)d_in[11], *c1r_bl = (const float*)d_in[12], *c1r_Wr = (const float*)d_in[13];
    const float* c2h_Wl = (const float*)d_in[14], *c2h_bl = (const float*)d_in[15], *c2h_Wr = (const float*)d_in[16];
    const float* c2r_Wl = (const float*)d_in[17], *c2r_bl = (const float*)d_in[18], *c2r_Wr = (const float*)d_in[19];
    float* out = (float*)d_out;

    // workspace layout (floats): aggT | aggP | invT | invP | t1 | p1
    float* aggT = (float*)d_ws;
    float* aggP = aggT + (size_t)HT * HH;
    float* invT = aggP + (size_t)HP * HH;
    float* invP = invT + (size_t)HT;
    float* t1   = invP + (size_t)HP;
    float* p1   = t1   + (size_t)HT * HH;

    const long aggFloats = (long)HT * HH + (long)HP * HH;           // contiguous agg region
    const long zAll4 = (aggFloats + HT + HP) / 4;                   // agg + deg region
    const long zAgg4 = aggFloats / 4;

    // 1) zero agg + degree buffers
    k_zero<<<(unsigned)((zAll4 + 255) / 256), 256, 0, stream>>>((float4*)aggT, zAll4);
    // 2) degree counts -> invT/invP buffers, then invert in place (contiguous T+P floats)
    k_degree<<<(HE + 255) / 256, 256, 0, stream>>>(esrc, edst, invP, invT, HE);
    k_invdeg<<<((HT + HP) + 255) / 256, 256, 0, stream>>>(invT, HT + HP);
    // 3) layer-1 neighbor sums
    k_scatter<<<(unsigned)((long)HE * 32 / 256), 256, 0, stream>>>(xp, xt, esrc, edst, aggT, aggP, HE);
    // 4) layer-1 SAGE + relu
    k_sage_gemm<<<HT / 16, 256, 0, stream>>>(aggT, invT, xt, c1h_Wl, c1h_bl, c1h_Wr, t1, 1);
    k_sage_gemm<<<HP / 16, 256, 0, stream>>>(aggP, invP, xp, c1r_Wl, c1r_bl, c1r_Wr, p1, 1);
    // 5) layer-2 neighbor sums
    k_zero<<<(unsigned)((zAgg4 + 255) / 256), 256, 0, stream>>>((float4*)aggT, zAgg4);
    k_scatter<<<(unsigned)((long)HE * 32 / 256), 256, 0, stream>>>(p1, t1, esrc, edst, aggT, aggP, HE);
    // 6) layer-2 SAGE (no relu), in-place over agg buffers -> t2 = aggT, p2 = aggP
    k_sage_gemm<<<HT / 16, 256, 0, stream>>>(aggT, invT, t1, c2h_Wl, c2h_bl, c2h_Wr, aggT, 0);
    k_sage_gemm<<<HP / 16, 256, 0, stream>>>(aggP, invP, p1, c2r_Wl, c2r_bl, c2r_Wr, aggP, 0);
    // 7) decoder
    k_dot<<<(unsigned)((long)HEL * 32 / 256), 256, 0, stream>>>(aggP, aggT, lsrc, ldst, out, HEL);
}